// GCN_10290741641786
// MI455X (gfx1250) — compile-verified
//
#include <hip/hip_runtime.h>
#include <hip/hip_bf16.h>

typedef float v2f __attribute__((ext_vector_type(2)));
typedef float v8f __attribute__((ext_vector_type(8)));

#define N_NODES_C 100000
#define D_C 128           // feature dim == channel dim
#define ROWS_PER_BLOCK 128
#define LDS_STRIDE 132    // 132 % 64 == 4 -> conflict-free b64 lane access

// LDS byte offset of a generic pointer that points into shared memory:
// explicit addrspacecast generic->AS3, then ptrtoint.
__device__ __forceinline__ unsigned lds_offset_of(const void* p) {
  return (unsigned)(size_t)(__attribute__((address_space(3))) const char*)p;
}

// GLOBAL_LOAD_ASYNC_TO_LDS_B128, GV mode: VDST = LDS byte address VGPR,
// VADDR = 64-bit global address pair, saddr = off. Tracked by ASYNCcnt.
__device__ __forceinline__ void async_load_b128(unsigned lds_addr, const void* gptr) {
  asm volatile("global_load_async_to_lds_b128 %0, %1, off"
               :
               : "v"(lds_addr), "v"((unsigned long long)(size_t)gptr)
               : "memory");
}

__device__ __forceinline__ void wait_asynccnt0() {
  asm volatile("s_wait_asynccnt 0" ::: "memory");
}

// -------------------- Kernel 1: h = X @ W via f32 WMMA --------------------
__global__ void gcn_gemm_xw(const float* __restrict__ X,
                            const float* __restrict__ W,
                            float* __restrict__ H) {
  extern __shared__ float smem[];
  float* Xs = smem;                                   // [128][132]
  float* Wt = smem + ROWS_PER_BLOCK * LDS_STRIDE;     // W transposed: Wt[n][k]

  const int tid     = threadIdx.x;
  const int rowBase = blockIdx.x * ROWS_PER_BLOCK;

  // Stage X tile: async global->LDS b128 (ASYNCcnt), no VGPR round-trip.
#pragma unroll
  for (int j = 0; j < 16; ++j) {
    int idx = tid + j * 256;              // float4 slot 0..4095
    int r   = idx >> 5;                   // 0..127
    int c4  = (idx & 31) << 2;            // 0..124
    int gr  = rowBase + r;
    float* dst = &Xs[r * LDS_STRIDE + c4];
    if (gr < N_NODES_C) {
      async_load_b128(lds_offset_of(dst), X + (size_t)gr * D_C + c4);
    } else {
      *(float4*)dst = make_float4(0.f, 0.f, 0.f, 0.f);
    }
  }

  // Stage W transposed (read coalesced row-major, scatter into LDS);
  // overlaps with the in-flight async X-tile transfers.
#pragma unroll 4
  for (int j = 0; j < 64; ++j) {
    int idx = tid + j * 256;              // 0..16383
    int k   = idx >> 7;                   // W row (K)
    int c   = idx & 127;                  // W col (N)
    Wt[c * LDS_STRIDE + k] = W[idx];
  }

  wait_asynccnt0();                       // my wave's async LDS writes landed
  __syncthreads();                        // publish to all waves in the block

  const int wave = tid >> 5;
  const int lane = tid & 31;
  const int ln   = lane & 15;             // N index (B/C/D) or M index (A)
  const int hi   = lane >> 4;             // K half-select for A/B fragments
  const int mr   = wave * 16;             // this wave's 16-row stripe in the tile

  v8f acc[8];
#pragma unroll
  for (int t = 0; t < 8; ++t) acc[t] = 0.f;

  // A frag: lanes 0-15 -> {K0,K1}, lanes 16-31 -> {K2,K3}  (ISA 7.12.2)
  const float* aBase = &Xs[(mr + ln) * LDS_STRIDE + 2 * hi];
  const float* bBase = &Wt[ln * LDS_STRIDE + 2 * hi];

#pragma unroll 4
  for (int kk = 0; kk < 32; ++kk) {
    v2f a = *(const v2f*)(aBase + 4 * kk);
#pragma unroll
    for (int t = 0; t < 8; ++t) {
      v2f b = *(const v2f*)(bBase + (size_t)(t * 16) * LDS_STRIDE + 4 * kk);
      acc[t] = __builtin_amdgcn_wmma_f32_16x16x4_f32(
          /*neg_a=*/false, a, /*neg_b=*/false, b,
          /*c_mod=*/(short)0, acc[t], /*reuse_a=*/false, /*reuse_b=*/false);
    }
  }

  // C/D layout: VGPR i -> row M=i (lanes 0-15) / M=8+i (lanes 16-31), N=ln
#pragma unroll
  for (int t = 0; t < 8; ++t) {
#pragma unroll
    for (int i = 0; i < 8; ++i) {
      int gr = rowBase + mr + i + 8 * hi;
      if (gr < N_NODES_C) H[(size_t)gr * D_C + t * 16 + ln] = acc[t][i];
    }
  }
}

// -------------------- Kernel 2: out = h*skip + bias (seed for atomics) ----
__global__ void gcn_seed_out(const float* __restrict__ H,
                             const float* __restrict__ skipw,
                             const float* __restrict__ bias,
                             float* __restrict__ out, int total) {
  int i = blockIdx.x * blockDim.x + threadIdx.x;
  if (i < total) {
    int c = i & (D_C - 1);
    out[i] = H[i] * skipw[c] + bias[c];
  }
}

// -------------------- Kernel 3: edge scatter-add (wave per edge) ----------
__global__ void gcn_scatter(const int* __restrict__ rows,
                            const int* __restrict__ cols,
                            const float* __restrict__ vals,
                            const float* __restrict__ H,
                            float* __restrict__ out, int nEdges) {
  int tid  = blockIdx.x * blockDim.x + threadIdx.x;
  int e    = tid >> 5;                    // whole wave shares one edge
  if (e >= nEdges) return;
  int lane = tid & 31;
  int r = rows[e];                        // wave-uniform -> scalar loads
  int c = cols[e];
  float v = vals[e];
  float4 m = *((const float4*)(H + (size_t)c * D_C) + lane);
  float* op = out + (size_t)r * D_C + lane * 4;
  unsafeAtomicAdd(op + 0, v * m.x);       // global_atomic_add_f32, no return
  unsafeAtomicAdd(op + 1, v * m.y);
  unsafeAtomicAdd(op + 2, v * m.z);
  unsafeAtomicAdd(op + 3, v * m.w);
}

// -------------------- Kernel 4: selu in place -----------------------------
__global__ void gcn_selu(float* __restrict__ out, int total) {
  int i = blockIdx.x * blockDim.x + threadIdx.x;
  if (i < total) {
    const float alpha = 1.6732632423543772f;
    const float scale = 1.0507009873554805f;
    float x = out[i];
    out[i] = x > 0.f ? scale * x : scale * alpha * (__expf(x) - 1.f);
  }
}

extern "C" void kernel_launch(void* const* d_in, const int* in_sizes, int n_in,
                              void* d_out, int out_size, void* d_ws, size_t ws_size,
                              hipStream_t stream) {
  const float* X     = (const float*)d_in[0];   // [N, 128]
  const int*   rows  = (const int*)  d_in[1];   // [E]
  const int*   cols  = (const int*)  d_in[2];   // [E]
  const float* vals  = (const float*)d_in[3];   // [E]
  const float* W     = (const float*)d_in[4];   // [128, 128]
  const float* bias  = (const float*)d_in[5];   // [128]
  const float* skipw = (const float*)d_in[6];   // [128]
  float* out = (float*)d_out;
  float* H   = (float*)d_ws;                    // h = X@W, 51.2 MB scratch

  const int nEdges = in_sizes[1];
  const int total  = N_NODES_C * D_C;

  const size_t shmem = (size_t)2 * ROWS_PER_BLOCK * LDS_STRIDE * sizeof(float); // 132 KB
  const int gemmBlocks = (N_NODES_C + ROWS_PER_BLOCK - 1) / ROWS_PER_BLOCK;

  gcn_gemm_xw<<<gemmBlocks, 256, shmem, stream>>>(X, W, H);
  gcn_seed_out<<<(total + 255) / 256, 256, 0, stream>>>(H, skipw, bias, out, total);
  long long sthreads = (long long)nEdges * 32;
  gcn_scatter<<<(int)((sthreads + 255) / 256), 256, 0, stream>>>(rows, cols, vals, H, out, nEdges);
  gcn_selu<<<(total + 255) / 256, 256, 0, stream>>>(out, total);
}